// MultiHeadAttention_74148315398272
// MI455X (gfx1250) — compile-verified
//
#include <hip/hip_runtime.h>
#include <cstddef>
#include <cstdint>

// ---------------------------------------------------------------------------
// Graph-Transformer layer for MI455X (gfx1250, wave32).
//   GEMMs (q,k,v,E_proj vs 128x128 weights) -> V_WMMA_F32_16X16X4_F32, with
//   A tiles streamed into LDS via GLOBAL_LOAD_ASYNC_TO_LDS_B128 (ASYNCcnt)
//   and double buffering. Edge phase: gathers (L2 resident) + f32 atomics.
// ---------------------------------------------------------------------------

typedef float v2f __attribute__((ext_vector_type(2)));
typedef float v8f __attribute__((ext_vector_type(8)));
typedef int   v4i __attribute__((__vector_size__(4 * sizeof(int))));

#define WSTR 132  // LDS row stride in floats (pad to spread banks)

#if defined(__AMDGCN__) && \
    __has_builtin(__builtin_amdgcn_global_load_async_to_lds_b128) && \
    __has_builtin(__builtin_amdgcn_s_wait_asynccnt)
#define ASYNC_LDS 1
typedef __attribute__((address_space(1))) v4i* gas_b128_t;  // global int4*
typedef __attribute__((address_space(3))) v4i* las_b128_t;  // LDS int4*
#else
#define ASYNC_LDS 0
#endif

#if ASYNC_LDS
// Issue async global->LDS copies for one 16x128 A tile (rows >= M skipped;
// their LDS rows stay stale, but D row m depends only on A row m and
// out-of-range rows are never stored).
__device__ __forceinline__ void stage_tile_async(const float* __restrict__ A,
                                                 int m0, int M,
                                                 float* ldsDst, int t) {
#pragma unroll
  for (int c = t; c < 512; c += 256) {  // 512 x 16B chunks, 2 per thread
    const int row = c >> 5;
    const int c16 = c & 31;
    if (m0 + row < M) {
      const float* gp = A + (size_t)(m0 + row) * 128 + c16 * 4;
      float* lp = ldsDst + row * WSTR + c16 * 4;
      __builtin_amdgcn_global_load_async_to_lds_b128(
          (gas_b128_t)gp, (las_b128_t)lp, 0, 0);
    }
  }
}
#endif

// C[M,128] = A[M,128] @ W[128,128], fp32 WMMA 16x16x4.
// Block = 256 threads = 8 waves; wave w owns output cols [16w,16w+16).
__launch_bounds__(256)
__global__ void gemm128_wmma_f32(const float* __restrict__ A,
                                 const float* __restrict__ W,
                                 float* __restrict__ C, int M) {
  __shared__ float WsT[128 * WSTR];   // transposed weights: WsT[n][k]
  __shared__ float As[2][16 * WSTR];  // double-buffered A tiles

  const int t = threadIdx.x;
  // Stage W transposed (coalesced global read, scattered LDS write).
  for (int i = t; i < 128 * 128; i += 256) {
    const int kk = i >> 7, nn = i & 127;
    WsT[nn * WSTR + kk] = W[i];
  }

  const int lane = t & 31;
  const int wave = t >> 5;
  const int half = lane >> 4;
  const int r16  = lane & 15;
  const int n0   = wave << 4;

  const int numTiles = (M + 15) >> 4;

#if ASYNC_LDS
  if (blockIdx.x < numTiles)
    stage_tile_async(A, blockIdx.x << 4, M, &As[0][0], t);
  int iter = 0;
  for (int tile = blockIdx.x; tile < numTiles; tile += gridDim.x, ++iter) {
    const int m0  = tile << 4;
    const int buf = iter & 1;
    __builtin_amdgcn_s_wait_asynccnt(0);
    __syncthreads();  // As[buf] visible to all; also covers W staging (iter 0)
    const int nxt = tile + gridDim.x;
    if (nxt < numTiles)
      stage_tile_async(A, nxt << 4, M, &As[buf ^ 1][0], t);
    const float* __restrict__ at = &As[buf][0];
#else
  for (int tile = blockIdx.x; tile < numTiles; tile += gridDim.x) {
    const int m0 = tile << 4;
    __syncthreads();
    for (int i = t; i < 16 * 128; i += 256) {
      const int r = i >> 7, c = i & 127;
      const int gr = m0 + r;
      As[0][r * WSTR + c] = (gr < M) ? A[(size_t)gr * 128 + c] : 0.0f;
    }
    __syncthreads();
    const float* __restrict__ at = &As[0][0];
#endif

    v8f acc = {};
    // ISA 7.12.2 fragment layout (16x16x4 f32):
    //  A: lanes 0-15 rows M=0..15; VGPR0 = K=0/K=2 per lane half, VGPR1 = K=1/K=3.
    //  B: VGPR0 = row K=0/K=2 per lane half, VGPR1 = K=1/K=3, col N = lane%16.
#pragma unroll
    for (int k = 0; k < 128; k += 4) {
      const int ka = k + 2 * half;
      v2f a, b;
      a.x = at[r16 * WSTR + ka];
      a.y = at[r16 * WSTR + ka + 1];
      b.x = WsT[(n0 + r16) * WSTR + ka];
      b.y = WsT[(n0 + r16) * WSTR + ka + 1];
      acc = __builtin_amdgcn_wmma_f32_16x16x4_f32(
          /*neg_a=*/false, a, /*neg_b=*/false, b,
          /*c_mod=*/(short)0, acc, /*reuse_a=*/false, /*reuse_b=*/false);
    }

    // D layout: VGPR i -> row m0+i (lanes 0-15) / m0+i+8 (lanes 16-31).
#pragma unroll
    for (int i = 0; i < 8; ++i) {
      const int gr = m0 + i + 8 * half;
      if (gr < M) C[(size_t)gr * 128 + n0 + r16] = acc[i];
    }
  }
}

// One thread per (edge, head): 16-wide triple-product dot, exp(clamp), then
// 16 f32 atomics into aV[dst] (L2 resident) + 1 into z[dst].
__launch_bounds__(256)
__global__ void edge_attn(const float* __restrict__ q, const float* __restrict__ k,
                          const float* __restrict__ v, const float* __restrict__ eproj,
                          const int* __restrict__ src, const int* __restrict__ dst,
                          float* __restrict__ aV, float* __restrict__ z, int E) {
  const int idx = blockIdx.x * blockDim.x + threadIdx.x;
  if (idx >= E * 8) return;
  const int eid = idx >> 3;
  const int hd  = idx & 7;
  const int s = src[eid];
  const int d = dst[eid];

  const float4* kp = (const float4*)(k + (size_t)s * 128 + hd * 16);
  const float4* qp = (const float4*)(q + (size_t)d * 128 + hd * 16);
  const float4* ep = (const float4*)(eproj + (size_t)eid * 128 + hd * 16);

  float dot = 0.0f;
#pragma unroll
  for (int j = 0; j < 4; ++j) {
    const float4 kk = kp[j], qq = qp[j], ee = ep[j];
    dot += kk.x * qq.x * ee.x + kk.y * qq.y * ee.y +
           kk.z * qq.z * ee.z + kk.w * qq.w * ee.w;
  }
  dot *= 0.25f;  // 1/sqrt(OUT_DIM=16)
  dot = fminf(fmaxf(dot, -5.0f), 5.0f);
  const float ss = expf(dot);

  const float4* vp = (const float4*)(v + (size_t)s * 128 + hd * 16);
  float* ap = aV + (size_t)d * 128 + hd * 16;
#pragma unroll
  for (int j = 0; j < 4; ++j) {
    const float4 vv = vp[j];
    atomicAdd(ap + 4 * j + 0, vv.x * ss);
    atomicAdd(ap + 4 * j + 1, vv.y * ss);
    atomicAdd(ap + 4 * j + 2, vv.z * ss);
    atomicAdd(ap + 4 * j + 3, vv.w * ss);
  }
  atomicAdd(z + (size_t)d * 8 + hd, ss);
}

__launch_bounds__(256)
__global__ void finalize_div(float* __restrict__ hout, const float* __restrict__ z, int N) {
  const int idx = blockIdx.x * blockDim.x + threadIdx.x;
  if (idx >= N * 128) return;
  const int n  = idx >> 7;
  const int hd = (idx >> 4) & 7;
  hout[idx] = hout[idx] / (z[n * 8 + hd] + 1e-6f);
}

__launch_bounds__(256)
__global__ void zero_f32(float* __restrict__ p, int n) {
  const int i = blockIdx.x * blockDim.x + threadIdx.x;
  if (i < n) p[i] = 0.0f;
}

extern "C" void kernel_launch(void* const* d_in, const int* in_sizes, int n_in,
                              void* d_out, int out_size, void* d_ws, size_t ws_size,
                              hipStream_t stream) {
  const float* h    = (const float*)d_in[0];
  const float* e    = (const float*)d_in[1];
  const float* h_in = (const float*)d_in[2];
  const float* Wq   = (const float*)d_in[3];
  const float* Wk   = (const float*)d_in[4];
  const float* Wv   = (const float*)d_in[5];
  const float* We   = (const float*)d_in[6];
  const int*   src  = (const int*)d_in[7];
  const int*   dst  = (const int*)d_in[8];

  const int N = in_sizes[0] / 128;  // 50000
  const int E = in_sizes[1] / 128;  // 800000

  float* out   = (float*)d_out;
  float* h_out = out;                    // N*128: aV accumulator, then result
  float* e_out = out + (size_t)N * 128;  // E*128: E_proj written here directly

  // Workspace: q, k, v projections + z accumulator (~78.4 MB).
  float* qws = (float*)d_ws;
  float* kws = qws + (size_t)N * 128;
  float* vws = kws + (size_t)N * 128;
  float* zws = vws + (size_t)N * 128;

  // Re-zero accumulators every call (atomic accumulation, poisoned buffers).
  {
    const int ta = N * 128;
    zero_f32<<<(ta + 255) / 256, 256, 0, stream>>>(h_out, ta);
    const int tz = N * 8;
    zero_f32<<<(tz + 255) / 256, 256, 0, stream>>>(zws, tz);
  }

  const int tilesN = (N + 15) >> 4;
  const int tilesE = (E + 15) >> 4;
  const int gN = tilesN < 2048 ? tilesN : 2048;
  const int gE = tilesE < 2048 ? tilesE : 2048;

  gemm128_wmma_f32<<<gN, 256, 0, stream>>>(h,    Wq, qws,   N);
  gemm128_wmma_f32<<<gN, 256, 0, stream>>>(h,    Wk, kws,   N);
  gemm128_wmma_f32<<<gN, 256, 0, stream>>>(h_in, Wv, vws,   N);
  gemm128_wmma_f32<<<gE, 256, 0, stream>>>(e,    We, e_out, E);

  const int totalEH = E * 8;
  edge_attn<<<(totalEH + 255) / 256, 256, 0, stream>>>(qws, kws, vws, e_out,
                                                       src, dst, h_out, zws, E);

  const int tf = N * 128;
  finalize_div<<<(tf + 255) / 256, 256, 0, stream>>>(h_out, zws, N);
}